// LMOptimizer_15023795601654
// MI455X (gfx1250) — compile-verified
//
#include <hip/hip_runtime.h>
#include <hip/hip_bf16.h>
#include <math.h>

// Geometry constants from the reference (B=16, H=W=512).
#define BATCH 16
#define HDIM  512
#define WDIM  512
#define PIX_PER_BLOCK 256
#define BLOCKS_PER_BATCH ((HDIM * WDIM) / PIX_PER_BLOCK)   // 1024
#define TOTAL_BLOCKS (BATCH * BLOCKS_PER_BATCH)            // 16384
#define BYTES_PER_BLOCK (PIX_PER_BLOCK * 9 * 4)            // 9216
#define CHUNKS_PER_BLOCK (BYTES_PER_BLOCK / 16)            // 576

__device__ __forceinline__ void async_store_lds_b128(unsigned long long gaddr,
                                                     unsigned lds_byte_off) {
    // CDNA5 async LDS->global store (VGLOBAL op 101, GV mode), ASYNCcnt-tracked.
    asm volatile("global_store_async_from_lds_b128 %0, %1, off"
                 :
                 : "v"(gaddr), "v"(lds_byte_off)
                 : "memory");
}

__device__ __forceinline__ void wait_asynccnt0() {
    asm volatile("s_wait_asynccnt 0" ::: "memory");
}

__global__ __launch_bounds__(PIX_PER_BLOCK) void jacobian_kernel(
    const float* __restrict__ roll,
    const float* __restrict__ pitch,
    const float* __restrict__ focal,
    float* __restrict__ out)
{
    __shared__ float lds[PIX_PER_BLOCK * 9];

    const int tid = threadIdx.x;
    const int blk = blockIdx.x;
    const int b   = blk >> 10;                         // / BLOCKS_PER_BATCH
    const int pix = ((blk & (BLOCKS_PER_BATCH - 1)) << 8) | tid;  // pixel in batch

    // Per-batch scalars (block-uniform -> scalar loads).
    const float r  = roll[b];
    const float pt = pitch[b];
    const float f  = focal[b];

    const float sr = __sinf(r),  cr = __cosf(r);
    const float sp = __sinf(pt), cp = __cosf(pt);

    // abc = (-sr*cp, -cr*cp, -sp)
    const float a0 = -sr * cp;
    const float a1 = -cr * cp;
    const float c3 = -sp;

    // J_rp (3x2): rows (-cr*cp, sr*sp), (sr*cp, cr*sp), (0, -cp)
    const float J00 = -cr * cp, J01 = sr * sp;
    const float J10 =  sr * cp, J11 = cr * sp;
    const float J21 = -cp;

    const int x = pix & (WDIM - 1);
    const int y = pix >> 9;

    const float invf = __builtin_amdgcn_rcpf(f);
    const float u = ((float)x + 0.5f - (float)(WDIM / 2)) * invf;
    const float v = ((float)y + 0.5f - (float)(HDIM / 2)) * invf;

    // ---- J_up (2x3) ------------------------------------------------------
    // proj = abc[:2] - c3*uv
    const float p0 = fmaf(-c3, u, a0);
    const float p1 = fmaf(-c3, v, a1);
    const float m2 = fmaf(p0, p0, p1 * p1);
    const float im  = __builtin_amdgcn_rsqf(fmaxf(m2, 1e-12f)); // 1/||proj|| (guarded)
    const float im3 = im * im * im;

    // A = J_proj2abc @ J_rp  (2x2)
    const float A00 = J00;
    const float A01 = fmaf(u, cp, J01);   // sr*sp + u*cp
    const float A10 = J10;
    const float A11 = fmaf(v, cp, J11);   // cr*sp + v*cp

    // g = J_proj2f = c3 * uv / f
    const float g0 = c3 * u * invf;
    const float g1 = c3 * v * invf;

    // (I/m - p p^T / m^3) applied to columns A[:,0], A[:,1], g
    const float t0 = fmaf(p0, A00, p1 * A10);
    const float t1 = fmaf(p0, A01, p1 * A11);
    const float tg = fmaf(p0, g0,  p1 * g1);

    const float D00 = A00 * im - p0 * t0 * im3;
    const float D01 = A01 * im - p0 * t1 * im3;
    const float D02 = g0  * im - p0 * tg * im3;
    const float D10 = A10 * im - p1 * t0 * im3;
    const float D11 = A11 * im - p1 * t1 * im3;
    const float D12 = g1  * im - p1 * tg * im3;

    // ---- J_lat (1x3) -----------------------------------------------------
    const float n1sq = fmaf(u, u, fmaf(v, v, 1.0f));
    const float in1  = __builtin_amdgcn_rsqf(n1sq);     // 1/||(u,v,1)||
    const float in13 = in1 * in1 * in1;

    // J_delta = (uv1/||uv1||) @ J_rp
    const float d0 = fmaf(u, J00, v * J10) * in1;
    const float d1 = fmaf(u, J01, fmaf(v, J11, J21)) * in1;

    // w = -uv/f ; J_f = (w.abc01)/n1 - (uv1.abc)*(uv.w)/n1^3
    const float w0  = -u * invf;
    const float w1  = -v * invf;
    const float svw = fmaf(u, w0, v * w1);
    const float vda = fmaf(u, a0, fmaf(v, a1, c3));
    const float wda = fmaf(w0, a0, w1 * a1);
    const float Jf  = wda * in1 - vda * svw * in13;

    // ---- stage into LDS in exact output layout (pixel-major, 9 f32/pixel)
    float* lp = &lds[tid * 9];
    lp[0] = D00; lp[1] = D01; lp[2] = D02;
    lp[3] = D10; lp[4] = D11; lp[5] = D12;
    lp[6] = d0;  lp[7] = d1;  lp[8] = Jf;

    __syncthreads();

    // ---- async LDS -> HBM streaming copy: contiguous b128 chunks ---------
    const unsigned long long gbase =
        (unsigned long long)out + (unsigned long long)blk * (unsigned long long)BYTES_PER_BLOCK;
    const unsigned lbase = (unsigned)(unsigned long long)(const void*)&lds[0];

#pragma unroll
    for (int i = 0; i < 3; ++i) {
        const int c = tid + i * PIX_PER_BLOCK;
        if (c < CHUNKS_PER_BLOCK) {
            async_store_lds_b128(gbase + (unsigned long long)(c * 16),
                                 lbase + (unsigned)(c * 16));
        }
    }
    wait_asynccnt0();
}

extern "C" void kernel_launch(void* const* d_in, const int* in_sizes, int n_in,
                              void* d_out, int out_size, void* d_ws, size_t ws_size,
                              hipStream_t stream) {
    (void)in_sizes; (void)n_in; (void)d_ws; (void)ws_size; (void)out_size;
    const float* roll  = (const float*)d_in[0];
    const float* pitch = (const float*)d_in[1];
    const float* focal = (const float*)d_in[2];
    float* out = (float*)d_out;

    hipLaunchKernelGGL(jacobian_kernel,
                       dim3(TOTAL_BLOCKS), dim3(PIX_PER_BLOCK), 0, stream,
                       roll, pitch, focal, out);
}